// CG_woFilter_cuda_37984690766237
// MI455X (gfx1250) — compile-verified
//
#include <hip/hip_runtime.h>

// ---------------------------------------------------------------------------
// Sparse Clebsch-Gordan tensor product (CGNet), lmax=5, tau=16, B=256.
// Per (l,l1,l2,m) the 16x16 (tau1 x tau2) complex output block is a
// 16x16xK fp32 GEMM (K = #valid m1, <= 11; complex-folded K' = 2K <= 22),
// computed with chains of V_WMMA_F32_16X16X4_F32 (two accumulators: Re, Im).
//
// v2: block = (8-batch octet) x (8-task group). Fs rows + entry tables are
// staged in LDS once; the inner loop is pure ds_load gathers + WMMA, and the
// only global traffic in steady state is the coalesced b64 output stream
// (240 MB total -> the 23.3 TB/s HBM roofline term, ~10 us).
// ---------------------------------------------------------------------------

#define LMAX   5
#define TAU    16
#define NTASKS 457            // sum over pairs of (2l+1)
#define TOT    116992         // output rows per batch
#define NB     256            // batch
#define FROW   1152           // 576 * 2 floats per batch row of Fs
#define MAXENT 24             // padded max entries per task (2K<=22 -> 24)
#define TGRP   8              // tasks per block
#define NGRP   ((NTASKS + TGRP - 1) / TGRP)   // 58

typedef float v2f __attribute__((ext_vector_type(2)));
typedef float v8f __attribute__((ext_vector_type(8)));

struct Entry {                // 20 bytes, align 4
    int   aR;                 // Fs offset (within batch row) for A_real source row
    int   aI;                 // Fs offset for A_imag source row
    int   b;                  // Fs offset for B source row
    float cR;                 // coefficient for A_real column (signed)
    float cI;                 // coefficient for A_imag column
};

struct Task {                 // 16 bytes
    int entry_start;          // index into entry array (== task_id * MAXENT)
    int n_chunks;             // number of K-chunks of 4
    int out_base;             // row offset within one batch's output
    int pad;
};

// ------------------------------ init kernel --------------------------------

__device__ static double dfact(int n) {
    double r = 1.0;
    for (int i = 2; i <= n; ++i) r *= (double)i;
    return r;
}

__device__ static double cg_coef(int j1, int m1, int j2, int m2, int j, int m) {
    if (m1 + m2 != m) return 0.0;
    int dj = j1 - j2; if (dj < 0) dj = -dj;
    if (j < dj || j > j1 + j2) return 0.0;
    double pref = sqrt((2.0 * j + 1.0) * dfact(j + j1 - j2) * dfact(j - j1 + j2) *
                       dfact(j1 + j2 - j) / dfact(j1 + j2 + j + 1));
    pref *= sqrt(dfact(j + m) * dfact(j - m) * dfact(j1 - m1) * dfact(j1 + m1) *
                 dfact(j2 - m2) * dfact(j2 + m2));
    int kmin = 0;
    if (j2 - j - m1 > kmin) kmin = j2 - j - m1;
    if (j1 + m2 - j > kmin) kmin = j1 + m2 - j;
    int kmax = j1 + j2 - j;
    if (j1 - m1 < kmax) kmax = j1 - m1;
    if (j2 + m2 < kmax) kmax = j2 + m2;
    double s = 0.0;
    for (int k = kmin; k <= kmax; ++k) {
        double d = dfact(k) * dfact(j1 + j2 - j - k) * dfact(j1 - m1 - k) *
                   dfact(j2 + m2 - k) * dfact(j - j2 + m1 + k) * dfact(j - j1 - m2 + k);
        s += ((k & 1) ? -1.0 : 1.0) / d;
    }
    return pref * s;
}

__global__ void cg_init_kernel(Task* __restrict__ tasks, Entry* __restrict__ entries) {
    if (threadIdx.x != 0 || blockIdx.x != 0) return;

    int cum[LMAX + 2];
    cum[0] = 0;
    for (int l = 0; l <= LMAX; ++l) cum[l + 1] = cum[l] + TAU * (2 * l + 1);

    int npl[LMAX + 1];
    for (int l = 0; l <= LMAX; ++l) {
        int c = 0;
        for (int l1 = 0; l1 <= LMAX; ++l1)
            for (int l2 = l1; l2 <= LMAX; ++l2) {
                int d = l2 - l1;
                if (d <= l && l <= l1 + l2) c++;
            }
        npl[l] = c;
    }
    int Lbase[LMAX + 1];
    int acc = 0;
    for (int l = 0; l <= LMAX; ++l) { Lbase[l] = acc; acc += (2 * l + 1) * 256 * npl[l]; }

    int task_id = 0;
    for (int l = 0; l <= LMAX; ++l) {
        int rank = 0;
        for (int l1 = 0; l1 <= LMAX; ++l1) {
            for (int l2 = l1; l2 <= LMAX; ++l2) {
                int d = l2 - l1;
                if (!(d <= l && l <= l1 + l2)) continue;
                for (int mi = 0; mi < 2 * l + 1; ++mi) {
                    int m = mi - l;
                    Entry* e = entries + task_id * MAXENT;
                    int n = 0;
                    // type-R entries (j < K): B row = r2, A_R = +cg*r1, A_I = +cg*i1
                    for (int m1 = -l1; m1 <= l1; ++m1) {
                        int m2 = m - m1;
                        if (m2 < -l2 || m2 > l2) continue;
                        float cg = (float)cg_coef(l1, m1, l2, m2, l, m);
                        int a_r = (cum[l1] + (m1 + l1) * TAU) * 2;
                        int b_r = (cum[l2] + (m2 + l2) * TAU) * 2;
                        e[n].aR = a_r;     e[n].cR = cg;
                        e[n].aI = a_r + 1; e[n].cI = cg;
                        e[n].b  = b_r;
                        n++;
                    }
                    // type-I entries (j >= K): B row = i2, A_R = -cg*i1, A_I = +cg*r1
                    for (int m1 = -l1; m1 <= l1; ++m1) {
                        int m2 = m - m1;
                        if (m2 < -l2 || m2 > l2) continue;
                        float cg = (float)cg_coef(l1, m1, l2, m2, l, m);
                        int a_r = (cum[l1] + (m1 + l1) * TAU) * 2;
                        int b_i = (cum[l2] + (m2 + l2) * TAU) * 2 + 1;
                        e[n].aR = a_r + 1; e[n].cR = -cg;
                        e[n].aI = a_r;     e[n].cI = cg;
                        e[n].b  = b_i;
                        n++;
                    }
                    int padded = (n + 3) & ~3;
                    for (; n < padded; ++n) {
                        e[n].aR = 0; e[n].aI = 0; e[n].b = 0;
                        e[n].cR = 0.0f; e[n].cI = 0.0f;
                    }
                    tasks[task_id].entry_start = task_id * MAXENT;
                    tasks[task_id].n_chunks   = padded >> 2;
                    tasks[task_id].out_base   = Lbase[l] + mi * 256 * npl[l] + rank * 256;
                    tasks[task_id].pad        = 0;
                    task_id++;
                }
                rank++;
            }
        }
    }
}

// ------------------------------ main kernel --------------------------------
// Block = (task group of 8) x (batch octet of 8). One wave32 per batch; each
// wave iterates the 8 tasks. All operand gathers come from LDS.
//
// A 16x4 f32 layout: lanes 0-15 hold M=lane, K=0 (v0), K=1 (v1);
//                    lanes 16-31 hold M=lane-16, K=2 (v0), K=3 (v1).
// B 4x16 f32 layout: v0 = rows K=0 (lanes 0-15, N=lane) / K=2 (lanes 16-31);
//                    v1 = rows K=1 / K=3.
// C/D 16x16 f32: VGPR g, lane L -> i = g + 8*(L>=16), j = L%16.

__global__ void __launch_bounds__(256)
cg_wmma_kernel(const float* __restrict__ Fs, float* __restrict__ out,
               const Task* __restrict__ tasks, const Entry* __restrict__ entries) {
    __shared__ float sF[8 * FROW];            // 36 KB: 8 batch rows
    __shared__ Entry sEnt[TGRP * MAXENT];     // 3.8 KB: entry tables for group

    const int tid    = threadIdx.x;
    const int grp    = blockIdx.x >> 5;       // task group (0..57)
    const int oct    = blockIdx.x & 31;       // batch octet (0..31)
    const int batch0 = oct * 8;
    int ntg = NTASKS - grp * TGRP;
    if (ntg > TGRP) ntg = TGRP;

    // stage 8 Fs rows (8 * 1152 floats = 2304 float4) cooperatively
    {
        const float4* __restrict__ src4 = (const float4*)(Fs + (size_t)batch0 * FROW);
        float4* dst4 = (float4*)sF;
        for (int i = tid; i < (8 * FROW) / 4; i += 256) dst4[i] = src4[i];
    }
    // stage entry tables for the task group (contiguous: entry_start = id*MAXENT)
    {
        const int ndw = ntg * MAXENT * 5;
        const int* __restrict__ esrc = (const int*)(entries + (size_t)grp * TGRP * MAXENT);
        int* edst = (int*)sEnt;
        for (int i = tid; i < ndw; i += 256) edst[i] = esrc[i];
    }
    __syncthreads();

    const int warp  = tid >> 5;
    const int lane  = tid & 31;
    const int m16   = lane & 15;              // tau index (M for A, N for B)
    const int kh    = (lane >> 4) << 1;       // 0 for lanes 0-15, 2 for 16-31
    const int ibase = (lane >> 4) * 8;
    const int batch = batch0 + warp;

    const float* __restrict__ fb = sF + warp * FROW;
    float* __restrict__ obase = out + (size_t)batch * (TOT * 2);

    for (int tt = 0; tt < ntg; ++tt) {
        const Task t = tasks[grp * TGRP + tt];
        const Entry* __restrict__ ent = sEnt + tt * MAXENT;

        v8f accR = {};
        v8f accI = {};
        for (int c = 0; c < t.n_chunks; ++c) {
            const Entry e0 = ent[c * 4 + kh];
            const Entry e1 = ent[c * 4 + kh + 1];

            v2f a_r, a_i, bv;
            a_r[0] = e0.cR * fb[e0.aR + m16 * 2];
            a_r[1] = e1.cR * fb[e1.aR + m16 * 2];
            a_i[0] = e0.cI * fb[e0.aI + m16 * 2];
            a_i[1] = e1.cI * fb[e1.aI + m16 * 2];
            bv[0]  = fb[e0.b + m16 * 2];
            bv[1]  = fb[e1.b + m16 * 2];

            accR = __builtin_amdgcn_wmma_f32_16x16x4_f32(
                       false, a_r, false, bv, (short)0, accR, false, false);
            accI = __builtin_amdgcn_wmma_f32_16x16x4_f32(
                       false, a_i, false, bv, (short)0, accI, false, false);
        }

        // write 8 (Re,Im) pairs per lane, coalesced 8-byte stores
        float* __restrict__ ob = obase + (size_t)t.out_base * 2;
#pragma unroll
        for (int g = 0; g < 8; ++g) {
            float2 v = make_float2(accR[g], accI[g]);
            *(float2*)(ob + (size_t)((ibase + g) * 16 + m16) * 2) = v;
        }
    }
}

// ------------------------------ launcher -----------------------------------

extern "C" void kernel_launch(void* const* d_in, const int* in_sizes, int n_in,
                              void* d_out, int out_size, void* d_ws, size_t ws_size,
                              hipStream_t stream) {
    const float* Fs = (const float*)d_in[0];
    float* out = (float*)d_out;

    Task*  tasks   = (Task*)d_ws;
    Entry* entries = (Entry*)((char*)d_ws + (size_t)NTASKS * sizeof(Task));

    // Build CG coefficient / entry tables on device (deterministic, capture-safe).
    cg_init_kernel<<<1, 32, 0, stream>>>(tasks, entries);

    // 58 task groups * 32 batch octets = 1856 blocks of 256 threads (8 waves).
    cg_wmma_kernel<<<NGRP * 32, 256, 0, stream>>>(Fs, out, tasks, entries);
}